// PrototypeAttentionLoss_52750788329575
// MI455X (gfx1250) — compile-verified
//
#include <hip/hip_runtime.h>
#include <math.h>

typedef float v2f __attribute__((ext_vector_type(2)));
typedef float v8f __attribute__((ext_vector_type(8)));

#define NB 64      // bands
#define ND 128     // embedding dim
#define NC 8       // classes
#define KC 256     // combined (r|i) dim
#define NBATCH 4096
#define EPSF 1e-8f
#define GMM_LAMBDA 0.01f

// LDS layout (floats); padded strides avoid bank conflicts; A operands are
// zero-padded to 16 rows so WMMA A-fragment loads need no EXEC masking.
#define ZC_STRIDE 264                          // zc[band][combined-dim], 64 rows
#define PC_STRIDE 264                          // pc[class(16)][combined-dim]
#define AL_STRIDE 68                           // al[class(16)][band]
#define OFF_ZC   0
#define OFF_PC   (OFF_ZC + NB * ZC_STRIDE)     // 16896, 16*264 = 4224
#define OFF_SC0  (OFF_PC + 16 * PC_STRIDE)     // 21120 : partial scores K-half 0
#define OFF_SC1  (OFF_SC0 + NC * 64)           // 21632 : partial scores K-half 1
#define OFF_AL   (OFF_SC1 + NC * 64)           // 22144, 16*68 = 1088
#define OFF_HAT  (OFF_AL + 16 * AL_STRIDE)     // 23232 : hat[8][256] (r|i)
#define OFF_RED  (OFF_HAT + NC * 256)          // 25280 : D2[8], D[8]
#define SMEM_FLOATS (OFF_RED + 16)             // 25296
#define SMEM_BYTES (SMEM_FLOATS * 4)           // 101184 B (< 320KB WGP LDS)

__global__ __launch_bounds__(256)
void proto_attn_loss_kernel(const float* __restrict__ Zr,
                            const float* __restrict__ Zi,
                            const float* __restrict__ proto,
                            const int*   __restrict__ labels,
                            float* __restrict__ out,   // [0]=loss, [1..4096]=preds, then alpha (B,C,Bn)
                            float* __restrict__ ws)    // [B] per-b loss partials
{
    extern __shared__ float smem[];
    const int tid  = threadIdx.x;
    const int b    = blockIdx.x;
    const int wave = tid >> 5;
    const int lane = tid & 31;
    const int lm   = lane & 15;   // M (or N) index inside a fragment
    const int kh   = lane >> 4;   // K-half select (0 -> K{0,1}, 1 -> K{2,3})

    // ---- Stage Zr[b], Zi[b] into LDS as zc[band][0..255] = [Zr | Zi] -------
    const float* zr = Zr + (size_t)b * NB * ND;
    const float* zi = Zi + (size_t)b * NB * ND;
#pragma unroll
    for (int i = 0; i < 8; ++i) {
        int flat = (tid + i * 256) * 4;           // 0..8188, multiple of 4
        int n = flat >> 7;                        // band
        int d = flat & 127;                       // dim (mult of 4)
        float4 vr = *(const float4*)(zr + flat);
        float4 vi = *(const float4*)(zi + flat);
        *(float4*)&smem[OFF_ZC + n * ZC_STRIDE + d]       = vr;
        *(float4*)&smem[OFF_ZC + n * ZC_STRIDE + 128 + d] = vi;
    }
    // proto -> pc[class][0..255] = [Pr | Pi]; rows 8..15 zero
#pragma unroll
    for (int i = 0; i < 8; ++i) {
        int j = tid + i * 256;                    // 0..2047
        int m = j >> 8, k = j & 255;
        float v = (k < 128) ? proto[(0 * NC + m) * ND + k]
                            : proto[(1 * NC + m) * ND + (k - 128)];
        smem[OFF_PC + m * PC_STRIDE + k] = v;
    }
    for (int i = tid; i < 8 * PC_STRIDE; i += 256) smem[OFF_PC + 8 * PC_STRIDE + i] = 0.0f;
    for (int i = tid; i < 8 * AL_STRIDE; i += 256) smem[OFF_AL + 8 * AL_STRIDE + i] = 0.0f;
    __syncthreads();

    // ---- GEMM 1: scores(16x64) = P(16x256) * Zc^T(256x64), split-K ---------
    // wave w: band tile (w&3), K-half (w>>2); partial tiles summed in softmax
    {
        const int t     = wave & 3;
        const int kbase = (wave >> 2) * 128;
        const int scb   = (wave >> 2) ? OFF_SC1 : OFF_SC0;
        v8f acc = {};
        for (int k0 = 0; k0 < 128; k0 += 4) {
            int ka = kbase + k0 + 2 * kh;
            v2f a, bf;
            a.x = smem[OFF_PC + lm * PC_STRIDE + ka];
            a.y = smem[OFF_PC + lm * PC_STRIDE + ka + 1];
            int band = t * 16 + lm;
            bf.x = smem[OFF_ZC + band * ZC_STRIDE + ka];
            bf.y = smem[OFF_ZC + band * ZC_STRIDE + ka + 1];
            acc = __builtin_amdgcn_wmma_f32_16x16x4_f32(false, a, false, bf,
                                                        (short)0, acc, false, false);
        }
        if (lane < 16) {                          // rows 0..7 = valid classes
            int n = t * 16 + lane;
#pragma unroll
            for (int j = 0; j < NC; ++j) smem[scb + j * 64 + n] = acc[j];
        }
    }
    __syncthreads();

    // ---- softmax over bands: wave c handles class c (64 vals, 2 per lane) --
    {
        const int c = wave;
        float s0 = smem[OFF_SC0 + c * 64 + lane]      + smem[OFF_SC1 + c * 64 + lane];
        float s1 = smem[OFF_SC0 + c * 64 + lane + 32] + smem[OFF_SC1 + c * 64 + lane + 32];
        float mx = fmaxf(s0, s1);
#pragma unroll
        for (int off = 16; off > 0; off >>= 1) mx = fmaxf(mx, __shfl_xor(mx, off, 32));
        float e0 = __expf(s0 - mx), e1 = __expf(s1 - mx);
        float sm = e0 + e1;
#pragma unroll
        for (int off = 16; off > 0; off >>= 1) sm += __shfl_xor(sm, off, 32);
        float inv = 1.0f / sm;
        float a0 = e0 * inv, a1 = e1 * inv;
        smem[OFF_AL + c * AL_STRIDE + lane]      = a0;
        smem[OFF_AL + c * AL_STRIDE + lane + 32] = a1;
        float* alpha_out = out + 1 + NBATCH + (size_t)b * NC * NB;
        alpha_out[c * NB + lane]      = a0;
        alpha_out[c * NB + lane + 32] = a1;
    }
    __syncthreads();

    // ---- GEMM 2: hat(16x256) = alpha(16x64) * Zc(64x256) -------------------
    // wave w owns dim-tiles w and w+8; A-fragment shared by both tiles
    {
        const int d0 = wave * 16 + lm;
        const int d1 = (wave + 8) * 16 + lm;
        v8f acc0 = {}, acc1 = {};
        for (int k0 = 0; k0 < NB; k0 += 4) {
            int ka = k0 + 2 * kh;
            v2f a, b0, b1;
            a.x = smem[OFF_AL + lm * AL_STRIDE + ka];
            a.y = smem[OFF_AL + lm * AL_STRIDE + ka + 1];
            b0.x = smem[OFF_ZC + ka * ZC_STRIDE + d0];
            b0.y = smem[OFF_ZC + (ka + 1) * ZC_STRIDE + d0];
            b1.x = smem[OFF_ZC + ka * ZC_STRIDE + d1];
            b1.y = smem[OFF_ZC + (ka + 1) * ZC_STRIDE + d1];
            acc0 = __builtin_amdgcn_wmma_f32_16x16x4_f32(false, a, false, b0,
                                                         (short)0, acc0, false, false);
            acc1 = __builtin_amdgcn_wmma_f32_16x16x4_f32(false, a, false, b1,
                                                         (short)0, acc1, false, false);
        }
        if (lane < 16) {
            int c0 = wave * 16 + lane, c1 = (wave + 8) * 16 + lane;
#pragma unroll
            for (int j = 0; j < NC; ++j) {
                smem[OFF_HAT + j * 256 + c0] = acc0[j];
                smem[OFF_HAT + j * 256 + c1] = acc1[j];
            }
        }
    }
    __syncthreads();

    // ---- epilogue: wave c reduces class c over d (4 dims per lane) ---------
    {
        const int c = wave;
        float d2 = 0.0f, Dacc = 0.0f;
#pragma unroll
        for (int i = 0; i < 4; ++i) {
            int d = lane + i * 32;
            float hr = smem[OFF_HAT + c * 256 + d];
            float hi = smem[OFF_HAT + c * 256 + 128 + d];
            float pr = smem[OFF_PC + c * PC_STRIDE + d];
            float pi = smem[OFF_PC + c * PC_STRIDE + 128 + d];
            float dr = hr - pr, di = hi - pi;
            d2 += dr * dr + di * di;
            float cross = fabsf(hr * pi - hi * pr);
            float magz  = sqrtf(hr * hr + hi * hi) + EPSF;
            float orth  = cross / magz;
            float dot   = hr * pr + hi * pi;
            float sgn   = (dot < 0.0f) ? (magz + (magz - orth)) : orth;
            float pmag  = sqrtf(pr * pr + pi * pi);
            Dacc += sgn + fabsf(magz - pmag);
        }
#pragma unroll
        for (int off = 16; off > 0; off >>= 1) {
            d2   += __shfl_xor(d2, off, 32);
            Dacc += __shfl_xor(Dacc, off, 32);
        }
        if (lane == 0) { smem[OFF_RED + c] = d2; smem[OFF_RED + 8 + c] = Dacc; }
    }
    __syncthreads();

    // ---- scalar tail: softmse, log-softmax CE, argmax ----------------------
    if (tid == 0) {
        float D2v[NC], Dv[NC], ex[NC];
#pragma unroll
        for (int c = 0; c < NC; ++c) { D2v[c] = smem[OFF_RED + c]; Dv[c] = smem[OFF_RED + 8 + c]; }
        // r = softmax(-D2); softmse = sum r*D2
        float m1 = -D2v[0];
#pragma unroll
        for (int c = 1; c < NC; ++c) m1 = fmaxf(m1, -D2v[c]);
        float se = 0.0f;
#pragma unroll
        for (int c = 0; c < NC; ++c) { ex[c] = __expf(-D2v[c] - m1); se += ex[c]; }
        float softmse = 0.0f;
#pragma unroll
        for (int c = 0; c < NC; ++c) softmse += (ex[c] / se) * D2v[c];
        // logits = -D; CE on label
        float m2 = -Dv[0];
#pragma unroll
        for (int c = 1; c < NC; ++c) m2 = fmaxf(m2, -Dv[c]);
        float s2 = 0.0f;
#pragma unroll
        for (int c = 0; c < NC; ++c) s2 += __expf(-Dv[c] - m2);
        float lse = m2 + __logf(s2);
        int lab = labels[b];
        float loss_cls = -((-Dv[lab]) - lse);
        ws[b] = loss_cls + GMM_LAMBDA * softmse;
        // pred = argmax(-D), first occurrence on ties
        int best = 0; float bv = -Dv[0];
#pragma unroll
        for (int c = 1; c < NC; ++c) { float v = -Dv[c]; if (v > bv) { bv = v; best = c; } }
        out[1 + b] = (float)best;
    }
}

__global__ __launch_bounds__(256)
void reduce_loss_kernel(const float* __restrict__ ws, float* __restrict__ out)
{
    __shared__ float s[256];
    int t = threadIdx.x;
    float acc = 0.0f;
    for (int i = t; i < NBATCH; i += 256) acc += ws[i];
    s[t] = acc;
    __syncthreads();
    for (int off = 128; off > 0; off >>= 1) {
        if (t < off) s[t] += s[t + off];
        __syncthreads();
    }
    if (t == 0) out[0] = s[0] / (float)NBATCH;
}

extern "C" void kernel_launch(void* const* d_in, const int* in_sizes, int n_in,
                              void* d_out, int out_size, void* d_ws, size_t ws_size,
                              hipStream_t stream) {
    (void)in_sizes; (void)n_in; (void)out_size; (void)ws_size;
    const float* Zr     = (const float*)d_in[0];
    const float* Zi     = (const float*)d_in[1];
    const float* proto  = (const float*)d_in[2];
    const int*   labels = (const int*)d_in[3];
    float* out = (float*)d_out;
    float* ws  = (float*)d_ws;

    // allow >64KB dynamic LDS (CDNA5 WGP has 320KB); harmless if already set
    (void)hipFuncSetAttribute((const void*)proto_attn_loss_kernel,
                              hipFuncAttributeMaxDynamicSharedMemorySize, SMEM_BYTES);

    proto_attn_loss_kernel<<<NBATCH, 256, SMEM_BYTES, stream>>>(Zr, Zi, proto, labels, out, ws);
    reduce_loss_kernel<<<1, 256, 0, stream>>>(ws, out);
}